// TheoryGuidedEdgeConstructor_58025008169383
// MI455X (gfx1250) — compile-verified
//
#include <hip/hip_runtime.h>
#include <hip/hip_bf16.h>
#include <math.h>

typedef float v2f __attribute__((ext_vector_type(2)));
typedef float v8f __attribute__((ext_vector_type(8)));

#define NROWS 8192
#define FDIM  64
#define HDIM  128
#define NNTOT 67108864ULL   // 8192*8192
#define TOPK  100
#define NBINS 4096
#define CAPC  (1u << 22)
#define LDP   132           // padded LDS row stride (floats): 132%64=4 banks/row -> conflict free

// ---------------- workspace layout (float-index offsets) ----------------
#define OFF_EMB   0u            // 8192*128
#define OFF_NORM  1048576u      // 8192
#define OFF_NSTD  1056768u      // 8192
#define OFF_CSUM  1064960u      // 64
#define OFF_CSQ   1065024u      // 64
#define OFF_MEAN  1065088u      // 64
#define OFF_STD0  1065152u      // 64
#define OFF_SVEC  1065216u      // 64
#define OFF_SCAL  1065280u      // 16: [0]=vol [1]=liq [3]=alpha [4]=beta [5]=gamma [6]=stress [7]=threshold
#define OFF_HIST  1065296u      // 4096 (as unsigned)
#define OFF_CNT   1069392u      // 1 (as unsigned)
#define OFF_CVAL  1069440u      // CAPC floats
#define OFF_CIDX  (1069440u + CAPC)  // CAPC ints

// ---------------- kernel 0: zero counters / hist / scalars ----------------
__global__ void zero_k(unsigned* hist, unsigned* cnt, float* scal) {
    int i = blockIdx.x * blockDim.x + threadIdx.x;
    if (i < NBINS) hist[i] = 0u;
    if (i == 0) *cnt = 0u;
    if (i < 16) scal[i] = 0.0f;
}

// ---------------- kernel 1: per-feature column sums ----------------
__global__ void colstats_k(const float* __restrict__ x, float* csum, float* csq) {
    __shared__ float rs[256], rq[256];
    int j = blockIdx.x;                // feature 0..63
    int t = threadIdx.x;
    float s = 0.f, q = 0.f;
    for (int i = t; i < NROWS; i += 256) {
        float v = x[i * FDIM + j];
        s += v; q += v * v;
    }
    rs[t] = s; rq[t] = q; __syncthreads();
    for (int o = 128; o > 0; o >>= 1) {
        if (t < o) { rs[t] += rs[t + o]; rq[t] += rq[t + o]; }
        __syncthreads();
    }
    if (t == 0) { csum[j] = rs[0]; csq[j] = rq[0]; }
}

// ---------------- kernel 2: mean0/std0, vol, liq ----------------
__global__ void tinyA_k(const float* csum, const float* csq,
                        float* mean0, float* std0, float* scal) {
    __shared__ float ra[64], rb[64];
    int j = threadIdx.x;  // 64 threads
    float m = csum[j] / (float)NROWS;
    float var = (csq[j] - (float)NROWS * m * m) / (float)(NROWS - 1);
    float sd = sqrtf(fmaxf(var, 0.0f));
    mean0[j] = m; std0[j] = sd;
    // global vol over all N*F elements
    ra[j] = csum[j]; rb[j] = csq[j]; __syncthreads();
    for (int o = 32; o > 0; o >>= 1) { if (j < o) { ra[j] += ra[j + o]; rb[j] += rb[j + o]; } __syncthreads(); }
    float S = ra[0], SS = rb[0];
    __syncthreads();
    // spread of std0 (ddof=1 over 64)
    ra[j] = sd; __syncthreads();
    for (int o = 32; o > 0; o >>= 1) { if (j < o) ra[j] += ra[j + o]; __syncthreads(); }
    float ms = ra[0] / 64.0f;
    __syncthreads();
    ra[j] = (sd - ms) * (sd - ms); __syncthreads();
    for (int o = 32; o > 0; o >>= 1) { if (j < o) ra[j] += ra[j + o]; __syncthreads(); }
    if (j == 0) {
        float M = (float)NROWS * (float)FDIM;
        float gm = S / M;
        float gv = (SS - M * gm * gm) / (M - 1.0f);
        scal[0] = fminf(sqrtf(fmaxf(gv, 0.0f)), 1.0f);           // vol
        float spread = sqrtf(ra[0] / 63.0f);
        scal[1] = fminf(1.0f / (1.0f + spread), 1.0f);           // liq
    }
}

// ---------------- kernel 3: s_j = sum_i (x - mean)/(std+eps) ----------------
__global__ void svec_k(const float* __restrict__ x, const float* mean0,
                       const float* std0, float* svec) {
    __shared__ float rs[256];
    int j = blockIdx.x;
    int t = threadIdx.x;
    float m = mean0[j], inv = 1.0f / (std0[j] + 1e-8f);
    float s = 0.f;
    for (int i = t; i < NROWS; i += 256) s += (x[i * FDIM + j] - m) * inv;
    rs[t] = s; __syncthreads();
    for (int o = 128; o > 0; o >>= 1) { if (t < o) rs[t] += rs[t + o]; __syncthreads(); }
    if (t == 0) svec[j] = rs[0];
}

// ---------------- kernel 4: corr, market MLP, alpha/beta/gamma/stress ----------------
__global__ void tinyB_k(const float* svec, float* scal,
                        const float* rp, const float* mW1, const float* mb1,
                        const float* mW2, const float* mb2, float* outf) {
    float ss = 0.f;
    for (int j = 0; j < FDIM; ++j) ss += svec[j] * svec[j];
    float corr = fminf(fmaxf(ss / 5.49755813888e11f, -1.0f), 1.0f);  // /N^3
    float market[3] = { scal[0], corr, scal[1] };
    float hid[32];
    for (int k = 0; k < 32; ++k) {
        float h = mb1[k];
        for (int i = 0; i < 3; ++i) h += market[i] * mW1[i * 32 + k];
        hid[k] = fmaxf(h, 0.0f);
    }
    float lg[3];
    for (int t = 0; t < 3; ++t) {
        float l = mb2[t];
        for (int k = 0; k < 32; ++k) l += hid[k] * mW2[k * 3 + t];
        lg[t] = l;
    }
    float mx = fmaxf(lg[0], fmaxf(lg[1], lg[2]));
    float e0 = __expf(lg[0] - mx), e1 = __expf(lg[1] - mx), e2 = __expf(lg[2] - mx);
    float den = e0 + e1 + e2;
    float p0 = e0 / den, p1 = e1 / den, p2 = e2 / den;
    float prm[3];
    for (int t = 0; t < 3; ++t)
        prm[t] = p0 * rp[0 * 3 + t] + p1 * rp[1 * 3 + t] + p2 * rp[2 * 3 + t];
    scal[3] = prm[0];  // alpha
    scal[4] = prm[1];  // beta
    scal[5] = prm[2];  // gamma
    scal[6] = market[0] * market[1] / (market[2] + 0.1f);  // stress
    outf[NNTOT + 300] = p0; outf[NNTOT + 301] = p1; outf[NNTOT + 302] = p2;
}

// ---------------- kernel 5: node encoder + norms + node_std ----------------
__global__ __launch_bounds__(HDIM) void encoder_k(const float* __restrict__ x,
        const float* __restrict__ W1, const float* __restrict__ b1,
        const float* __restrict__ W2, const float* __restrict__ b2,
        float* __restrict__ emb, float* __restrict__ norms, float* __restrict__ nstd) {
    __shared__ float xr[FDIM], hb[HDIM], red[HDIM];
    int row = blockIdx.x;
    int j = threadIdx.x;
    if (j < FDIM) xr[j] = x[row * FDIM + j];
    __syncthreads();
    float h = b1[j];
    #pragma unroll 8
    for (int f = 0; f < FDIM; ++f) h += xr[f] * W1[f * HDIM + j];
    hb[j] = fmaxf(h, 0.0f);
    __syncthreads();
    float e = b2[j];
    #pragma unroll 8
    for (int k = 0; k < HDIM; ++k) e += hb[k] * W2[k * HDIM + j];
    emb[row * HDIM + j] = e;
    red[j] = e * e; __syncthreads();
    for (int o = 64; o > 0; o >>= 1) { if (j < o) red[j] += red[j + o]; __syncthreads(); }
    if (j == 0) norms[row] = fmaxf(sqrtf(red[0]), 1e-8f);
    __syncthreads();
    red[j] = (j < FDIM) ? xr[j] : 0.0f; __syncthreads();
    for (int o = 64; o > 0; o >>= 1) { if (j < o) red[j] += red[j + o]; __syncthreads(); }
    float mean = red[0] / (float)FDIM;
    __syncthreads();
    red[j] = (j < FDIM) ? (xr[j] - mean) * (xr[j] - mean) : 0.0f; __syncthreads();
    for (int o = 64; o > 0; o >>= 1) { if (j < o) red[j] += red[j + o]; __syncthreads(); }
    if (j == 0) nstd[row] = sqrtf(red[0] / (float)(FDIM - 1));
}

// ---------------- kernel 6: WMMA GEMM + fused sigmoid epilogue ----------------
// Staging uses CDNA5 async global->LDS DMA (ASYNCcnt-tracked), bypassing VGPRs.
__global__ __launch_bounds__(256) void gemm_k(const float* __restrict__ emb,
        const float* __restrict__ norms, const float* __restrict__ nstd,
        const float* __restrict__ scal, float* __restrict__ probs) {
    extern __shared__ float smem[];
    float* tI = smem;                  // 128 x LDP
    float* tJ = tI + 128 * LDP;        // 128 x LDP
    float* nI = tJ + 128 * LDP;        // 128
    float* nJ = nI + 128;
    float* sI = nJ + 128;
    float* sJ = sI + 128;

    const int tid = threadIdx.x;
    const int iBase = blockIdx.y * 128;
    const int jBase = blockIdx.x * 128;

    // stage emb tiles: 16 iters x (256 threads * 16B) x2 = 2 x 128x128 floats
    // global_load_async_to_lds_b128: copies 16B from global addr vaddr to LDS addr vdst; both 16B aligned
    // (LDS row pitch 528B = 33*16, global row pitch 512B).
    for (int it = 0; it < 16; ++it) {
        int row = it * 8 + (tid >> 5);
        int col = (tid & 31) * 4;
        unsigned li = (unsigned)(uintptr_t)(tI + row * LDP + col);
        unsigned lj = (unsigned)(uintptr_t)(tJ + row * LDP + col);
        unsigned long long gi = (unsigned long long)(uintptr_t)(emb + (size_t)(iBase + row) * HDIM + col);
        unsigned long long gj = (unsigned long long)(uintptr_t)(emb + (size_t)(jBase + row) * HDIM + col);
        asm volatile("global_load_async_to_lds_b128 %0, %1, off"
                     :: "v"(li), "v"(gi) : "memory");
        asm volatile("global_load_async_to_lds_b128 %0, %1, off"
                     :: "v"(lj), "v"(gj) : "memory");
    }
    if (tid < 128) {
        nI[tid] = norms[iBase + tid]; nJ[tid] = norms[jBase + tid];
        sI[tid] = nstd[iBase + tid];  sJ[tid] = nstd[jBase + tid];
    }
    asm volatile("s_wait_asynccnt 0" ::: "memory");
    __syncthreads();

    const int wave = tid >> 5;      // 0..7 -> 16-row strip
    const int lane = tid & 31;
    const int lm = lane & 15;       // M (for A) / N (for B)
    const int lh = lane >> 4;       // K-half selector

    v8f acc[8];
    #pragma unroll
    for (int jt = 0; jt < 8; ++jt) acc[jt] = (v8f){};

    const float* aRow = tI + (wave * 16 + lm) * LDP + 2 * lh;
    #pragma unroll 4
    for (int k0 = 0; k0 < HDIM; k0 += 4) {
        v2f a = *(const v2f*)(aRow + k0);
        #pragma unroll
        for (int jt = 0; jt < 8; ++jt) {
            v2f b = *(const v2f*)(tJ + (jt * 16 + lm) * LDP + k0 + 2 * lh);
            acc[jt] = __builtin_amdgcn_wmma_f32_16x16x4_f32(
                false, a, false, b, (short)0, acc[jt], false, false);
        }
    }

    const float alpha = scal[3], beta = scal[4], gamma = scal[5], stress = scal[6];
    const float bs = beta * stress;
    #pragma unroll
    for (int jt = 0; jt < 8; ++jt) {
        int gj = jBase + jt * 16 + lm;
        float invnj = 1.0f / nJ[jt * 16 + lm];
        float sj = sJ[jt * 16 + lm];
        #pragma unroll
        for (int v = 0; v < 8; ++v) {
            int m = v + 8 * lh;
            int gi = iBase + wave * 16 + m;
            float sim = acc[jt][v] * invnj / nI[wave * 16 + m];
            float asym = fabsf(sI[wave * 16 + m] - sj);
            float logit = alpha * sim + bs - gamma * asym;
            float p = (gi == gj) ? 0.0f : 1.0f / (1.0f + __expf(-logit));
            // streaming 256MB output: keep it out of L2 (emb tiles live there)
            __builtin_nontemporal_store(p, probs + (size_t)gi * NROWS + gj);
        }
    }
}

// ---------------- kernel 7: histogram of probs ----------------
__global__ void hist_k(const float* __restrict__ probs, unsigned* __restrict__ hist) {
    __shared__ unsigned lh[NBINS];
    int t = threadIdx.x;
    for (int b = t; b < NBINS; b += 256) lh[b] = 0u;
    __syncthreads();
    size_t stride = (size_t)gridDim.x * 256;
    for (size_t i = (size_t)blockIdx.x * 256 + t; i < NNTOT; i += stride) {
        float p = __builtin_nontemporal_load(probs + i);
        int b = (int)(p * (float)NBINS);
        b = b < 0 ? 0 : (b > NBINS - 1 ? NBINS - 1 : b);
        atomicAdd(&lh[b], 1u);
    }
    __syncthreads();
    for (int b = t; b < NBINS; b += 256)
        if (lh[b]) atomicAdd(&hist[b], lh[b]);
}

// ---------------- kernel 8: find threshold bin for top-K ----------------
__global__ void thresh_k(const unsigned* hist, float* scal) {
    unsigned cum = 0;
    for (int b = NBINS - 1; b >= 0; --b) {
        cum += hist[b];
        if (cum >= TOPK) { scal[7] = (float)b / (float)NBINS; return; }
    }
    scal[7] = 0.0f;
}

// ---------------- kernel 9: gather candidates >= threshold ----------------
__global__ void gather_k(const float* __restrict__ probs, const float* scal,
                         unsigned* cnt, float* cval, int* cidx) {
    float thr = scal[7];
    size_t stride = (size_t)gridDim.x * blockDim.x;
    for (size_t i = (size_t)blockIdx.x * blockDim.x + threadIdx.x; i < NNTOT; i += stride) {
        float p = __builtin_nontemporal_load(probs + i);
        if (p >= thr) {
            unsigned pos = atomicAdd(cnt, 1u);
            if (pos < CAPC) { cval[pos] = p; cidx[pos] = (int)i; }
        }
    }
}

// ---------------- kernel 10: select top-100 (deterministic by (val desc, idx asc)) ----------------
__global__ __launch_bounds__(1024) void select_k(const unsigned* cnt, float* cval,
                                                 const int* cidx, float* outf, int* outi) {
    __shared__ float sv[1024];
    __shared__ int   si[1024];
    __shared__ int   sp[1024];
    int t = threadIdx.x;
    unsigned n = *cnt; if (n > CAPC) n = CAPC;
    for (int k = 0; k < TOPK; ++k) {
        float bv = -1.0f; int bi = 0x7fffffff; int bp = -1;
        for (unsigned j = t; j < n; j += 1024) {
            float v = cval[j];
            int   id = cidx[j];
            if (v > bv || (v == bv && id < bi)) { bv = v; bi = id; bp = (int)j; }
        }
        sv[t] = bv; si[t] = bi; sp[t] = bp;
        __syncthreads();
        for (int o = 512; o > 0; o >>= 1) {
            if (t < o) {
                float v2 = sv[t + o]; int i2 = si[t + o];
                if (v2 > sv[t] || (v2 == sv[t] && i2 < si[t])) {
                    sv[t] = v2; si[t] = i2; sp[t] = sp[t + o];
                }
            }
            __syncthreads();
        }
        if (t == 0) {
            if (sv[0] >= 0.0f) {
                outf[NNTOT + k] = sv[0];
                outi[NNTOT + TOPK + k]     = si[0] / NROWS;
                outi[NNTOT + 2 * TOPK + k] = si[0] % NROWS;
                cval[sp[0]] = -1.0f;   // remove selected
            } else {
                outf[NNTOT + k] = 0.0f;
                outi[NNTOT + TOPK + k] = 0;
                outi[NNTOT + 2 * TOPK + k] = 0;
            }
        }
        __threadfence_block();
        __syncthreads();
    }
}

// ---------------- host launcher ----------------
extern "C" void kernel_launch(void* const* d_in, const int* in_sizes, int n_in,
                              void* d_out, int out_size, void* d_ws, size_t ws_size,
                              hipStream_t stream) {
    const float* x   = (const float*)d_in[0];
    const float* W1  = (const float*)d_in[1];
    const float* b1  = (const float*)d_in[2];
    const float* W2  = (const float*)d_in[3];
    const float* b2  = (const float*)d_in[4];
    const float* rp  = (const float*)d_in[5];
    const float* mW1 = (const float*)d_in[6];
    const float* mb1 = (const float*)d_in[7];
    const float* mW2 = (const float*)d_in[8];
    const float* mb2 = (const float*)d_in[9];

    float* ws = (float*)d_ws;
    float* emb   = ws + OFF_EMB;
    float* norms = ws + OFF_NORM;
    float* nstd  = ws + OFF_NSTD;
    float* csum  = ws + OFF_CSUM;
    float* csq   = ws + OFF_CSQ;
    float* mean0 = ws + OFF_MEAN;
    float* std0  = ws + OFF_STD0;
    float* svec  = ws + OFF_SVEC;
    float* scal  = ws + OFF_SCAL;
    unsigned* hist = (unsigned*)(ws + OFF_HIST);
    unsigned* cnt  = (unsigned*)(ws + OFF_CNT);
    float* cval = ws + OFF_CVAL;
    int*   cidx = (int*)(ws + OFF_CIDX);

    float* outf = (float*)d_out;
    int*   outi = (int*)d_out;

    zero_k<<<16, 256, 0, stream>>>(hist, cnt, scal);
    colstats_k<<<FDIM, 256, 0, stream>>>(x, csum, csq);
    tinyA_k<<<1, 64, 0, stream>>>(csum, csq, mean0, std0, scal);
    svec_k<<<FDIM, 256, 0, stream>>>(x, mean0, std0, svec);
    tinyB_k<<<1, 1, 0, stream>>>(svec, scal, rp, mW1, mb1, mW2, mb2, outf);
    encoder_k<<<NROWS, HDIM, 0, stream>>>(x, W1, b1, W2, b2, emb, norms, nstd);

    size_t smemBytes = (size_t)(2 * 128 * LDP + 4 * 128) * sizeof(float);  // 137,216 B
    dim3 gg(NROWS / 128, NROWS / 128);
    gemm_k<<<gg, 256, smemBytes, stream>>>(emb, norms, nstd, scal, outf);

    hist_k<<<4096, 256, 0, stream>>>(outf, hist);
    thresh_k<<<1, 1, 0, stream>>>(hist, scal);
    gather_k<<<4096, 256, 0, stream>>>(outf, scal, cnt, cval, cidx);
    select_k<<<1, 1024, 0, stream>>>(cnt, cval, cidx, outf, outi);
}